// CrossTransformer_36756330119370
// MI455X (gfx1250) — compile-verified
//
#include <hip/hip_runtime.h>

typedef __attribute__((ext_vector_type(2))) float v2f;
typedef __attribute__((ext_vector_type(4))) float v4f;
typedef __attribute__((ext_vector_type(8))) float v8f;

#define B_DIM 64
#define C_DIM 256
#define T_DIM 120
#define V_DIM 25
#define BT_DIM (B_DIM * T_DIM)          // 7680
#define TOTAL  (B_DIM * C_DIM * T_DIM * V_DIM)  // 49,152,000
#define LDSTRIDE 260                     // 256 + 4 pad: conflict-free b64 LDS loads

// ---------------------------------------------------------------------------
// Kernel 1: O[bt][c] = Wo @ (Wv @ y_vec[bt] + bv) + bo, fp32 WMMA 16x16x4.
// One workgroup = 16 bt rows, 8 waves, each wave owns two 16-wide N tiles.
// ---------------------------------------------------------------------------
__global__ __launch_bounds__(256)
void ct_gemm_chain(const float* __restrict__ y,    // [B, C, T]
                   const float* __restrict__ Wv,   // [C, C]
                   const float* __restrict__ bv,   // [C]
                   const float* __restrict__ Wo,   // [C, C]
                   const float* __restrict__ bo,   // [C]
                   float* __restrict__ O)          // [BT, C]
{
    __shared__ float sY[16 * LDSTRIDE];   // Y block  [m=16][k=256]
    __shared__ float sT[16 * LDSTRIDE];   // GEMM1 out [m=16][k=256]

    const int bt0 = blockIdx.x * 16;
    const int tid = threadIdx.x;

    // Stage Y block: row m -> bt = bt0+m; col c -> y[b*C*T + c*T + t]
    for (int i = tid; i < 16 * C_DIM; i += 256) {
        const int m  = i >> 8;
        const int c  = i & 255;
        const int bt = bt0 + m;
        const int b  = bt / T_DIM;
        const int t  = bt - b * T_DIM;
        sY[m * LDSTRIDE + c] = y[(b * C_DIM + c) * T_DIM + t];
    }
    __syncthreads();

    const int wave = tid >> 5;          // 0..7
    const int lane = tid & 31;
    const int half = lane >> 4;         // 0: K={0,1}, 1: K={2,3} per WMMA frag
    const int lm   = lane & 15;

    // ---- GEMM1: sT = sY @ Wv^T + bv ------------------------------------
    #pragma unroll
    for (int tile = 0; tile < 2; ++tile) {
        const int n = wave * 32 + tile * 16 + lm;
        const float* wrow = Wv + n * C_DIM;           // B[k][n] = Wv[n][k]
        v8f acc = {};
        for (int k0 = 0; k0 < C_DIM; k0 += 4) {
            v2f a = *(const v2f*)&sY[lm * LDSTRIDE + k0 + 2 * half];
            v2f b = *(const v2f*)&wrow[k0 + 2 * half];
            acc = __builtin_amdgcn_wmma_f32_16x16x4_f32(
                      false, a, false, b, (short)0, acc, false, false);
        }
        const float bias = bv[n];
        #pragma unroll
        for (int i = 0; i < 8; ++i)                   // D: m = i + 8*half, col n
            sT[(i + 8 * half) * LDSTRIDE + n] = acc[i] + bias;
    }
    __syncthreads();

    // ---- GEMM2: O = sT @ Wo^T + bo -------------------------------------
    #pragma unroll
    for (int tile = 0; tile < 2; ++tile) {
        const int n = wave * 32 + tile * 16 + lm;
        const float* wrow = Wo + n * C_DIM;
        v8f acc = {};
        for (int k0 = 0; k0 < C_DIM; k0 += 4) {
            v2f a = *(const v2f*)&sT[lm * LDSTRIDE + k0 + 2 * half];
            v2f b = *(const v2f*)&wrow[k0 + 2 * half];
            acc = __builtin_amdgcn_wmma_f32_16x16x4_f32(
                      false, a, false, b, (short)0, acc, false, false);
        }
        const float bias = bo[n];
        #pragma unroll
        for (int i = 0; i < 8; ++i)
            O[(bt0 + i + 8 * half) * C_DIM + n] = acc[i] + bias;
    }
}

// ---------------------------------------------------------------------------
// Kernel 2: z[b,c,t,v] = x[b,c,t,v] + O[(b*T+t)*C + c]
// float4 per thread; T*V = 3000 is divisible by 4 so a chunk never crosses a
// (b,c) plane — only t (div 25) can change inside the chunk. ~393 MB stream.
// ---------------------------------------------------------------------------
__global__ __launch_bounds__(256)
void ct_bias_add(const float* __restrict__ x,
                 const float* __restrict__ O,
                 float* __restrict__ z)
{
    const int i  = blockIdx.x * 256 + threadIdx.x;
    const int e0 = i * 4;
    if (e0 >= TOTAL) return;

    const int plane = e0 / (T_DIM * V_DIM);          // = b*C + c
    const int r     = e0 - plane * (T_DIM * V_DIM);  // in-plane offset
    const int b     = plane >> 8;
    const int c     = plane & 255;

    const v4f xv = *(const v4f*)(x + e0);
    const float* obase = O + (b * T_DIM) * C_DIM + c;

    v4f ov;
    #pragma unroll
    for (int j = 0; j < 4; ++j) {
        const int t = (r + j) / V_DIM;               // mul-shift, cheap
        ov[j] = obase[t * C_DIM];
    }
    *(v4f*)(z + e0) = xv + ov;
}

// ---------------------------------------------------------------------------
extern "C" void kernel_launch(void* const* d_in, const int* in_sizes, int n_in,
                              void* d_out, int out_size, void* d_ws, size_t ws_size,
                              hipStream_t stream) {
    const float* x  = (const float*)d_in[0];
    const float* y  = (const float*)d_in[1];
    // d_in[2..5] = Wq, bq, Wk, bk: mathematically dead (entmax over a
    // singleton axis is identically 1.0, so attention reduces to broadcast).
    const float* Wv = (const float*)d_in[6];
    const float* bv = (const float*)d_in[7];
    const float* Wo = (const float*)d_in[8];
    const float* bo = (const float*)d_in[9];
    float* z = (float*)d_out;
    float* O = (float*)d_ws;                  // BT*C*4 = 7.86 MB scratch

    ct_gemm_chain<<<BT_DIM / 16, 256, 0, stream>>>(y, Wv, bv, Wo, bo, O);
    ct_bias_add<<<(TOTAL / 4 + 255) / 256, 256, 0, stream>>>(x, O, z);
}